// CrossEntropyLossWeight3_1211180778080
// MI455X (gfx1250) — compile-verified
//
#include <hip/hip_runtime.h>
#include <math.h>

typedef float v2f __attribute__((ext_vector_type(2)));
typedef float v4f __attribute__((ext_vector_type(4)));
typedef float v8f __attribute__((ext_vector_type(8)));

#define B_ROWS (1 << 21)   // 2097152 rows
#define NCLS   16
#define BLOCK  256
#define RPT    8
#define GRID   (B_ROWS / (BLOCK * RPT))   // 1024 blocks

// Reduce 256 f32 partials held in LDS using V_WMMA_F32_16X16X4_F32 with an
// all-ones B matrix. Must be called by a fully-active wave32 (tid < 32).
// A-matrix layout (ISA 7.12.2, 32-bit A 16x4): lanes 0-15 hold K=0,1 of row
// M=lane; lanes 16-31 hold K=2,3 of row M=lane-16. Four chained K=4 tiles
// cover all 256 values; D[m][n] holds row-group sums replicated across N, so
// summing the 8 D VGPRs per lane gives M=0..7 (lanes 0-15) / M=8..15
// (lanes 16-31); one shfl_xor(16) completes the sum.
__device__ __forceinline__ float wmma_reduce256(const float* s, int lane) {
  const int m    = lane & 15;
  const int half = lane >> 4;
  v2f b; b[0] = 1.0f; b[1] = 1.0f;   // B = ones(4x16)
  v8f acc = {};
#pragma unroll
  for (int t = 0; t < 4; ++t) {
    const float* base = s + t * 64 + m * 4 + half * 2;
    v2f a; a[0] = base[0]; a[1] = base[1];
    acc = __builtin_amdgcn_wmma_f32_16x16x4_f32(
        /*neg_a=*/false, a, /*neg_b=*/false, b,
        /*c_mod=*/(short)0, acc, /*reuse_a=*/false, /*reuse_b=*/false);
  }
  float r = ((acc[0] + acc[1]) + (acc[2] + acc[3])) +
            ((acc[4] + acc[5]) + (acc[6] + acc[7]));
  r += __shfl_xor(r, 16, 32);
  return r;  // full 256-element sum, valid in every lane
}

__global__ __launch_bounds__(BLOCK)
void cew3_partial(const float* __restrict__ predict,
                  const float* __restrict__ target,
                  const float* __restrict__ penalty,
                  float* __restrict__ partials) {
  __shared__ float s_pen[NCLS * NCLS];
  __shared__ float s_red[BLOCK];
  const int tid = threadIdx.x;
  s_pen[tid] = penalty[tid];           // 256 == 16x16, one load per thread
  __syncthreads();

  const v4f* __restrict__ p4 = reinterpret_cast<const v4f*>(predict);
  const v4f* __restrict__ t4 = reinterpret_cast<const v4f*>(target);

  float acc = 0.0f;
#pragma unroll
  for (int r = 0; r < RPT; ++r) {
    const long long row =
        (long long)blockIdx.x * (BLOCK * RPT) + (long long)r * BLOCK + tid;

    // ---- predict: max + argmax (first occurrence), keep values for exp ----
    v4f pv[4];
    float m = -__builtin_inff();
    int pre = 0;
#pragma unroll
    for (int c = 0; c < 4; ++c) {
      pv[c] = __builtin_nontemporal_load(p4 + row * 4 + c);  // b128 NT stream
#pragma unroll
      for (int j = 0; j < 4; ++j) {
        float x = pv[c][j];
        if (x > m) { m = x; pre = c * 4 + j; }
      }
    }

    // ---- target: argmax only (streamed, nothing kept) ----
    float tm = -__builtin_inff();
    int tar = 0;
#pragma unroll
    for (int c = 0; c < 4; ++c) {
      v4f tv = __builtin_nontemporal_load(t4 + row * 4 + c);
#pragma unroll
      for (int j = 0; j < 4; ++j) {
        float x = tv[j];
        if (x > tm) { tm = x; tar = c * 4 + j; }
      }
    }

    // softmax prob at the predicted class == 1 / sum(exp(x - max)),
    // because exp(predict[pre] - max) == exp(0) == 1 exactly.
    float sum = 0.0f;
#pragma unroll
    for (int c = 0; c < 4; ++c)
#pragma unroll
      for (int j = 0; j < 4; ++j) sum += __expf(pv[c][j] - m);

    const float w = (pre == tar) ? 0.0f : s_pen[tar * NCLS + pre];
    // sum >= 1.0 and finite -> hardware v_rcp_f32 (1 ULP) is safe; avoids the
    // v_div_scale/v_div_fmas/v_div_fixup sequence seen in the previous build.
    acc += w * __builtin_amdgcn_rcpf(sum);
  }

  s_red[tid] = acc;
  __syncthreads();
  if (tid < 32) {                       // wave 0, EXEC all-ones for WMMA
    float s = wmma_reduce256(s_red, tid);
    if (tid == 0) partials[blockIdx.x] = s;
  }
}

__global__ __launch_bounds__(BLOCK)
void cew3_final(const float* __restrict__ partials, float* __restrict__ out) {
  __shared__ float s_red[BLOCK];
  const int tid = threadIdx.x;
  float a = partials[tid] + partials[tid + 256] +
            partials[tid + 512] + partials[tid + 768];   // GRID == 1024
  s_red[tid] = a;
  __syncthreads();
  if (tid < 32) {
    float s = wmma_reduce256(s_red, tid);
    if (tid == 0) out[0] = s * (1.0f / (float)B_ROWS);   // exact pow2 scale
  }
}

extern "C" void kernel_launch(void* const* d_in, const int* in_sizes, int n_in,
                              void* d_out, int out_size, void* d_ws, size_t ws_size,
                              hipStream_t stream) {
  const float* predict = (const float*)d_in[0];   // [B, 16] f32
  const float* target  = (const float*)d_in[1];   // [B, 16] f32
  const float* penalty = (const float*)d_in[2];   // [16, 16] f32
  float* out      = (float*)d_out;                // scalar f32
  float* partials = (float*)d_ws;                 // GRID floats (4 KB)

  cew3_partial<<<GRID, BLOCK, 0, stream>>>(predict, target, penalty, partials);
  cew3_final<<<1, BLOCK, 0, stream>>>(partials, out);
}